// PAM_11364483465723
// MI455X (gfx1250) — compile-verified
//
#include <hip/hip_runtime.h>
#include <hip/hip_bf16.h>

// Self-attention (PAM) for B=4, H=W=64, C=64, d=8 on gfx1250 (wave32).
// f32 WMMA flash-attention + double-buffered GLOBAL_LOAD_ASYNC_TO_LDS staging.

typedef float v2f __attribute__((ext_vector_type(2)));
typedef float v4f __attribute__((ext_vector_type(4)));
typedef float v8f __attribute__((ext_vector_type(8)));
typedef int   v4i __attribute__((ext_vector_type(4)));

#define AS1 __attribute__((address_space(1)))
#define AS3 __attribute__((address_space(3)))

#define BATCH 4
#define NTOK  4096     // 64*64 tokens
#define CDIM  64
#define DQK   8
#define KTILE 64       // keys per pipeline stage

#if __has_builtin(__builtin_amdgcn_global_load_async_to_lds_b128)
#define PAM_ASYNC 1
#else
#define PAM_ASYNC 0
#endif

static __device__ __forceinline__ void copy16_to_lds(const float* g, float* l) {
#if PAM_ASYNC
    // async DMA-style copy, tracked by ASYNCcnt, no VGPR round trip.
    // Builtin signature (per hipcc diagnostic): (int4 AS1*, int4 AS3*, imm offset, imm cpol)
    __builtin_amdgcn_global_load_async_to_lds_b128(
        (AS1 v4i*)(AS1 void*)(void*)g,
        (AS3 v4i*)(AS3 void*)l, 0, 0);
#else
    *(v4f*)l = *(const v4f*)g;
#endif
}

static __device__ __forceinline__ void wait_async_done() {
#if PAM_ASYNC
#if __has_builtin(__builtin_amdgcn_s_wait_asynccnt)
    __builtin_amdgcn_s_wait_asynccnt(0);
#else
    asm volatile("s_wait_asynccnt 0x0" ::: "memory");
#endif
#endif
}

static __device__ __forceinline__ v8f wmma_f32(v2f a, v2f b, v8f c) {
    // D = A(16x4) * B(4x16) + C(16x16), all f32, wave32
    return __builtin_amdgcn_wmma_f32_16x16x4_f32(false, a, false, b, (short)0, c, false, false);
}

static __device__ __forceinline__ float redmax16(float v) {
    v = fmaxf(v, __shfl_xor(v, 1, 32));
    v = fmaxf(v, __shfl_xor(v, 2, 32));
    v = fmaxf(v, __shfl_xor(v, 4, 32));
    v = fmaxf(v, __shfl_xor(v, 8, 32));
    return v;  // max over the 16-lane half-group (one C-layout row)
}

static __device__ __forceinline__ float redsum16(float v) {
    v += __shfl_xor(v, 1, 32);
    v += __shfl_xor(v, 2, 32);
    v += __shfl_xor(v, 4, 32);
    v += __shfl_xor(v, 8, 32);
    return v;
}

// ---------------------------------------------------------------------------
// Kernel 1: QK/V projection.  Per block: one 128-token tile; 8 waves each do a
// 16-token tile.  QK packs q (cols 0..7) and k (cols 8..15) into one 16-wide
// output so a single WMMA chain produces both.
// ---------------------------------------------------------------------------
__global__ __launch_bounds__(256) void pam_proj_kernel(
    const float* __restrict__ x, const float* __restrict__ Wq,
    const float* __restrict__ Wk, const float* __restrict__ Wv,
    float* __restrict__ qkbuf, float* __restrict__ vbuf)
{
    __shared__ float sX[128][66];   // padded: avoids half-wave bank conflicts
    __shared__ float sWqk[64][16];
    __shared__ float sWv[64][80];   // padded stride 80 -> conflict-free B loads

    const int tid  = threadIdx.x;
    const int lane = tid & 31;
    const int wave = tid >> 5;
    const int b    = blockIdx.x >> 5;   // 32 tiles of 128 tokens per batch
    const int tok0 = (blockIdx.x & 31) * 128;

    const float* xb = x + ((size_t)b * NTOK + tok0) * CDIM;
    for (int i = tid; i < 128 * CDIM; i += 256)
        sX[i >> 6][i & 63] = xb[i];
    for (int i = tid; i < CDIM * DQK; i += 256) {
        sWqk[i >> 3][(i & 7)]     = Wq[i];
        sWqk[i >> 3][(i & 7) + 8] = Wk[i];
    }
    for (int i = tid; i < CDIM * CDIM; i += 256)
        sWv[i >> 6][i & 63] = Wv[i];
    __syncthreads();

    const int h = lane >> 4;      // half-wave selects K-pair {0,1} vs {2,3}
    const int n = lane & 15;      // A: row M;  B/C: column N
    const int mrow = wave * 16 + n;

    v8f accqk = {};
    v8f accv[4] = {{}, {}, {}, {}};
#pragma unroll
    for (int c = 0; c < 16; ++c) {          // K = 64 in steps of 4
        const int kb = 4 * c + 2 * h;
        v2f a;  a.x = sX[mrow][kb];   a.y = sX[mrow][kb + 1];
        v2f bq; bq.x = sWqk[kb][n];   bq.y = sWqk[kb + 1][n];
        accqk = wmma_f32(a, bq, accqk);
#pragma unroll
        for (int t = 0; t < 4; ++t) {
            v2f bv; bv.x = sWv[kb][16 * t + n]; bv.y = sWv[kb + 1][16 * t + n];
            accv[t] = wmma_f32(a, bv, accv[t]);
        }
    }

    // C-layout store: VGPR r holds row r (lanes 0-15) / row r+8 (lanes 16-31)
#pragma unroll
    for (int r = 0; r < 8; ++r) {
        const int row = r + 8 * h;
        const size_t tok = (size_t)b * NTOK + tok0 + wave * 16 + row;
        qkbuf[tok * 16 + n] = accqk[r];
#pragma unroll
        for (int t = 0; t < 4; ++t)
            vbuf[tok * CDIM + 16 * t + n] = accv[t][r];
    }
}

// ---------------------------------------------------------------------------
// Kernel 2: flash attention.  Per block: 128 query rows (8 waves x 16 rows).
// 64 pipeline stages of 64 keys, double-buffered in LDS via async copies that
// overlap the 72 WMMAs of the current stage.
// ---------------------------------------------------------------------------
__global__ __launch_bounds__(256) void pam_attn_kernel(
    const float* __restrict__ qkbuf, const float* __restrict__ vbuf,
    const float* __restrict__ x, const float* __restrict__ gamma,
    float* __restrict__ out)
{
    __shared__ float sK[2][KTILE][12];   // 64 keys x 8 feats, stride 48B (16B-aligned rows)
    __shared__ float sV[2][KTILE][80];   // 64 keys x 64 feats, stride 320B
    __shared__ float sP[8][16][20];      // per-wave C->A relayout scratch

    const int tid  = threadIdx.x;
    const int lane = tid & 31;
    const int wave = tid >> 5;
    const int b    = blockIdx.x >> 5;
    const int q0   = (blockIdx.x & 31) * 128 + wave * 16;

    const int h = lane >> 4;
    const int n = lane & 15;

    // Q fragment in A-layout: d=8 -> two K=4 chunks
    const size_t qtok = (size_t)b * NTOK + q0 + n;
    v2f aq[2];
#pragma unroll
    for (int c = 0; c < 2; ++c) {
        const int kb = 4 * c + 2 * h;
        aq[c].x = qkbuf[qtok * 16 + kb];
        aq[c].y = qkbuf[qtok * 16 + kb + 1];
    }

    auto load_stage = [&](int bb, int k0) {
        const size_t base = (size_t)b * NTOK + k0;
        for (int q = tid; q < KTILE * 16; q += 256) {        // V: 1024 x 16B chunks
            const int row = q >> 4, c4 = (q & 15) << 2;
            copy16_to_lds(&vbuf[(base + row) * CDIM + c4], &sV[bb][row][c4]);
        }
        if (tid < KTILE * 2) {                               // K: 128 x 16B chunks
            const int row = tid >> 1, f4 = (tid & 1) << 2;
            copy16_to_lds(&qkbuf[(base + row) * 16 + 8 + f4], &sK[bb][row][f4]);
        }
    };

    v8f o[4] = {{}, {}, {}, {}};
    float mrow[8], lrow[8];
#pragma unroll
    for (int r = 0; r < 8; ++r) { mrow[r] = -__builtin_inff(); lrow[r] = 0.0f; }

    load_stage(0, 0);
    wait_async_done();
    __syncthreads();

    const int NITER = NTOK / KTILE;   // 64
    for (int j = 0; j < NITER; ++j) {
        const int bb = j & 1;
        if (j + 1 < NITER) load_stage(bb ^ 1, (j + 1) * KTILE);   // overlaps compute

        // S = Q * K^T : four 16x16 subtiles over 64 keys
        v8f s[4] = {{}, {}, {}, {}};
#pragma unroll
        for (int kt = 0; kt < 4; ++kt) {
#pragma unroll
            for (int c = 0; c < 2; ++c) {
                const int kb = 4 * c + 2 * h;
                v2f bk; bk.x = sK[bb][16 * kt + n][kb]; bk.y = sK[bb][16 * kt + n][kb + 1];
                s[kt] = wmma_f32(aq[c], bk, s[kt]);
            }
        }

        // Online softmax across the 64-key stage
        float corr[8];
#pragma unroll
        for (int r = 0; r < 8; ++r) {
            float tm = fmaxf(fmaxf(s[0][r], s[1][r]), fmaxf(s[2][r], s[3][r]));
            tm = redmax16(tm);
            const float mn = fmaxf(mrow[r], tm);
            corr[r] = __expf(mrow[r] - mn);
            float psum = 0.0f;
#pragma unroll
            for (int kt = 0; kt < 4; ++kt) {
                s[kt][r] = __expf(s[kt][r] - mn);
                psum += s[kt][r];
            }
            lrow[r] = lrow[r] * corr[r] + redsum16(psum);
            mrow[r] = mn;
        }
#pragma unroll
        for (int t = 0; t < 4; ++t)
#pragma unroll
            for (int r = 0; r < 8; ++r) o[t][r] *= corr[r];

        // O += P * V  (per subtile: C-layout -> A-layout via wave-private LDS)
#pragma unroll
        for (int kt = 0; kt < 4; ++kt) {
#pragma unroll
            for (int r = 0; r < 8; ++r) sP[wave][r + 8 * h][n] = s[kt][r];
#pragma unroll
            for (int c = 0; c < 4; ++c) {
                const int kb = 4 * c + 2 * h;
                v2f pa; pa.x = sP[wave][n][kb]; pa.y = sP[wave][n][kb + 1];
#pragma unroll
                for (int t = 0; t < 4; ++t) {
                    v2f bv;
                    bv.x = sV[bb][16 * kt + kb][16 * t + n];
                    bv.y = sV[bb][16 * kt + kb + 1][16 * t + n];
                    o[t] = wmma_f32(pa, bv, o[t]);
                }
            }
        }

        wait_async_done();     // next stage's tiles have landed
        __syncthreads();       // all waves done reading buf[bb] before reuse
    }

    // Epilogue: out = gamma * (O / l) + x
    const float g = gamma[0];
#pragma unroll
    for (int r = 0; r < 8; ++r) {
        const int row = r + 8 * h;
        const size_t tok = (size_t)b * NTOK + q0 + row;
        const float inv = 1.0f / lrow[r];
#pragma unroll
        for (int t = 0; t < 4; ++t) {
            const size_t idx = tok * CDIM + 16 * t + n;
            out[idx] = g * (o[t][r] * inv) + x[idx];
        }
    }
}

extern "C" void kernel_launch(void* const* d_in, const int* in_sizes, int n_in,
                              void* d_out, int out_size, void* d_ws, size_t ws_size,
                              hipStream_t stream) {
    (void)in_sizes; (void)n_in; (void)out_size; (void)ws_size;
    const float* x     = (const float*)d_in[0];
    const float* Wq    = (const float*)d_in[1];
    const float* Wk    = (const float*)d_in[2];
    const float* Wv    = (const float*)d_in[3];
    const float* gamma = (const float*)d_in[4];
    float* out = (float*)d_out;

    float* qkbuf = (float*)d_ws;                                  // B*N*16 f32 (1 MB)
    float* vbuf  = qkbuf + (size_t)BATCH * NTOK * 16;             // B*N*64 f32 (4 MB)

    dim3 grid(BATCH * (NTOK / 128));   // 128 blocks
    dim3 block(256);                   // 8 waves (wave32)
    pam_proj_kernel<<<grid, block, 0, stream>>>(x, Wq, Wk, Wv, qkbuf, vbuf);
    pam_attn_kernel<<<grid, block, 0, stream>>>(qkbuf, vbuf, x, gamma, out);
}